// PulseLoss_23510650978790
// MI455X (gfx1250) — compile-verified
//
#include <hip/hip_runtime.h>
#include <math.h>

// ---------------------------------------------------------------------------
// PulseLoss for MI455X (gfx1250, wave32).
//
// g = clip(mask @ K, 0, 1) with K[t,s] = exp(-(s-t)^2/8) is a banded matmul
// (f32 kernel underflows to 0.0f beyond |s-t| ~ 29). One wave owns a
// 16(batch) x 16(position) G-tile and accumulates 20 K=4 steps of
// V_WMMA_F32_16X16X4_F32 over the band t in [s0-32, s0+48).
//
// The Gaussian B-matrix is tile-invariant (dt = kk - 32 - n, independent of
// s0), so it is computed once into registers before the MAC chain. A-matrix
// (mask) pairs are loaded as unconditional 8B-aligned b64 loads with
// branch-free edge masking, all issued before the WMMA chain so loads batch.
//
// ISA VGPR layouts used (CDNA5 ISA, 32-bit operands, wave32):
//   A 16x4 : lane l, VGPR j  -> A[l%16][2*(l>=16)+j]
//   B 4x16 : lane l, VGPR j  -> B[2*(l>=16)+j][l%16]
//   C/D    : lane l, VGPR r  -> D[r + 8*(l>=16)][l%16]
//
// Reductions: fixed-order partial sums in d_ws (no float atomics ->
// deterministic), finalize writes (total, bce, count).
// ---------------------------------------------------------------------------

typedef float v2f __attribute__((ext_vector_type(2)));
typedef float v8f __attribute__((ext_vector_type(8)));

#define SEQ_LEN   8192
#define NBATCH    32
#define BAND_LO   32              // band starts at s0 - 32
#define KSTEPS    20              // band width 80 = 20 * K4
#define NTILE_S   (SEQ_LEN / 16)  // 512 position tiles

// d_ws layout (floats)
#define WS_BCE 0     // [NTILE_S] per-block BCE partial sums
#define WS_CP  512   // [NBATCH]  count of pred > 0.5 per batch
#define WS_CT  544   // [NBATCH]  sum of target per batch

__global__ __launch_bounds__(64)
void pulse_bce_wmma(const float* __restrict__ pred,
                    const float* __restrict__ target,
                    float* __restrict__ ws)
{
    const int s0        = blockIdx.x * 16;       // position tile base
    const int lane      = threadIdx.x & 31;
    const int wave      = threadIdx.x >> 5;      // 0: batches 0-15, 1: 16-31
    const int batchBase = wave * 16;
    const int halfSel   = lane >> 4;             // K-pair select (0,1)/(2,3)
    const int ln        = lane & 15;             // A-row M and B/C/D-col N

    // --- B matrix: tile-invariant Gaussian, computed once into registers ---
    v2f bmat[KSTEPS];
#pragma unroll
    for (int ks = 0; ks < KSTEPS; ++ks) {
#pragma unroll
        for (int j = 0; j < 2; ++j) {
            const float dt = (float)(ks * 4 + halfSel * 2 + j - BAND_LO - ln);
            bmat[ks][j] = expf(-0.125f * dt * dt);   // K[t, s] band value
        }
    }

    // --- A matrix: mask pairs, unconditional aligned b64 loads, edge-masked ---
    const float* __restrict__ row = target + (batchBase + ln) * SEQ_LEN;
    const int t0 = s0 - BAND_LO;                 // even (s0 mult of 16)
    v2f amat[KSTEPS];
#pragma unroll
    for (int ks = 0; ks < KSTEPS; ++ks) {
        const int pt  = t0 + ks * 4 + halfSel * 2;         // even index
        int ptc = pt;
        if (ptc < 0) ptc = 0;
        if (ptc > SEQ_LEN - 2) ptc = SEQ_LEN - 2;          // stays even
        const v2f tv = *reinterpret_cast<const v2f*>(row + ptc);
        const bool v0 = (pt     >= 0) && (pt     < SEQ_LEN);
        const bool v1 = (pt + 1 >= 0) && (pt + 1 < SEQ_LEN);
        amat[ks][0] = (v0 && tv[0] > 0.0f) ? 1.0f : 0.0f;
        amat[ks][1] = (v1 && tv[1] > 0.0f) ? 1.0f : 0.0f;
    }

    // --- Back-to-back WMMA MAC chain (full f32) ---
    v8f acc = {0.0f, 0.0f, 0.0f, 0.0f, 0.0f, 0.0f, 0.0f, 0.0f};
#pragma unroll
    for (int ks = 0; ks < KSTEPS; ++ks) {
        acc = __builtin_amdgcn_wmma_f32_16x16x4_f32(
            /*neg_a=*/false, amat[ks], /*neg_b=*/false, bmat[ks],
            /*c_mod=*/(short)0, acc, /*reuse_a=*/false, /*reuse_b=*/false);
    }

    // --- Per-lane: 8 (batch, position) elements of G tile -> BCE terms ---
    float bce_local = 0.0f;
#pragma unroll
    for (int r = 0; r < 8; ++r) {
        float g = acc[r];
        g = fminf(fmaxf(g, 0.0f), 1.0f);                   // clip(g, 0, 1)
        const int b = batchBase + r + 8 * halfSel;
        const float p   = pred[b * SEQ_LEN + s0 + ln];
        const float lp  = fmaxf(logf(p),    -100.0f);      // torch log clamp
        const float l1p = fmaxf(log1pf(-p), -100.0f);
        bce_local -= g * lp + (1.0f - g) * l1p;
    }

    // --- Fixed-order block reduction -> one workspace slot per block ---
    __shared__ float red[64];
    red[threadIdx.x] = bce_local;
    __syncthreads();
#pragma unroll
    for (int off = 32; off > 0; off >>= 1) {
        if (threadIdx.x < off) red[threadIdx.x] += red[threadIdx.x + off];
        __syncthreads();
    }
    if (threadIdx.x == 0) ws[WS_BCE + blockIdx.x] = red[0];
}

__global__ __launch_bounds__(256)
void pulse_count(const float* __restrict__ pred,
                 const float* __restrict__ target,
                 float* __restrict__ ws)
{
    const int b = blockIdx.x;                    // one block per batch
    float cp = 0.0f, ct = 0.0f;
    for (int t = threadIdx.x; t < SEQ_LEN; t += 256) {
        cp += (pred[b * SEQ_LEN + t] > 0.5f) ? 1.0f : 0.0f;
        ct += target[b * SEQ_LEN + t];
    }
    __shared__ float r0[256];
    __shared__ float r1[256];
    r0[threadIdx.x] = cp;
    r1[threadIdx.x] = ct;
    __syncthreads();
#pragma unroll
    for (int off = 128; off > 0; off >>= 1) {
        if (threadIdx.x < off) {
            r0[threadIdx.x] += r0[threadIdx.x + off];
            r1[threadIdx.x] += r1[threadIdx.x + off];
        }
        __syncthreads();
    }
    if (threadIdx.x == 0) {
        ws[WS_CP + b] = r0[0];
        ws[WS_CT + b] = r1[0];
    }
}

__global__ __launch_bounds__(64)
void pulse_finalize(const float* __restrict__ ws, float* __restrict__ out)
{
    __shared__ float red[64];
    float s = 0.0f;
    for (int i = threadIdx.x; i < NTILE_S; i += 64) s += ws[WS_BCE + i];
    red[threadIdx.x] = s;
    __syncthreads();
#pragma unroll
    for (int off = 32; off > 0; off >>= 1) {
        if (threadIdx.x < off) red[threadIdx.x] += red[threadIdx.x + off];
        __syncthreads();
    }
    if (threadIdx.x == 0) {
        const float bce = red[0] / (float)(NBATCH * SEQ_LEN);
        float cacc = 0.0f;
        for (int b = 0; b < NBATCH; ++b)
            cacc += fabsf(ws[WS_CP + b] - ws[WS_CT + b]);
        const float count_loss = 0.1f * (cacc / (float)NBATCH);
        out[0] = bce + count_loss;   // total_loss
        out[1] = bce;                // bce_loss
        out[2] = count_loss;         // count_loss
    }
}

extern "C" void kernel_launch(void* const* d_in, const int* in_sizes, int n_in,
                              void* d_out, int out_size, void* d_ws, size_t ws_size,
                              hipStream_t stream)
{
    const float* pred   = (const float*)d_in[0];  // [32, 8192, 1] f32
    const float* target = (const float*)d_in[1];  // [32, 8192, 1] f32
    float* ws  = (float*)d_ws;
    float* out = (float*)d_out;                   // 3 f32 scalars

    // All ws slots are fully written before being read; no init pass needed.
    pulse_bce_wmma<<<NTILE_S, 64, 0, stream>>>(pred, target, ws);
    pulse_count  <<<NBATCH, 256, 0, stream>>>(pred, target, ws);
    pulse_finalize<<<1, 64, 0, stream>>>(ws, out);
}